// PairwiseEuclideanLoss_81681688035493
// MI455X (gfx1250) — compile-verified
//
#include <hip/hip_runtime.h>
#include <hip/hip_bf16.h>

// Pairwise squared-distance mean via closed form:
//   total = n * sum_i ||x_i||^2 - || sum_i x_i ||^2
//   loss  = (total/2) / (n*(n-1)/2)
//
// Phase 1: streaming reduction (HBM-bound, ~134MB @ 23.3TB/s ~ 5.8us).
// Phase 2: WMMA-based cross-block column-sum reduction + finalize.

#define D_COLS 2048
#define TPB    512   // 512 threads * float4 = 2048 columns per block

typedef __attribute__((ext_vector_type(2))) float v2f;
typedef __attribute__((ext_vector_type(8))) float v8f;

// ---------------------------------------------------------------------------
// Phase 1: each block grid-strides over rows; per iteration the whole block
// reads one contiguous 8KB row with b128 loads. Thread t owns columns
// 4t..4t+3. Writes per-block column partials + per-block sqsum partial.
// ---------------------------------------------------------------------------
__global__ void __launch_bounds__(TPB)
pel_colsum_sq_kernel(const float* __restrict__ X,
                     float* __restrict__ colPart,   // [nblk][2048]
                     float* __restrict__ sqPart,    // [nblk]
                     int nrows, int nblk) {
    const int tid  = threadIdx.x;
    const int col4 = tid * 4;

    float4 cs = make_float4(0.f, 0.f, 0.f, 0.f);
    float  sq = 0.f;

    #pragma unroll 4
    for (int r = blockIdx.x; r < nrows; r += nblk) {
        const float4 v =
            *reinterpret_cast<const float4*>(X + (size_t)r * D_COLS + col4);
        cs.x += v.x; cs.y += v.y; cs.z += v.z; cs.w += v.w;
        sq += v.x * v.x + v.y * v.y + v.z * v.z + v.w * v.w;
    }

    reinterpret_cast<float4*>(colPart + (size_t)blockIdx.x * D_COLS)[tid] = cs;

    __shared__ float red[TPB];
    red[tid] = sq;
    __syncthreads();
    for (int s = TPB / 2; s > 0; s >>= 1) {
        if (tid < s) red[tid] += red[tid + s];
        __syncthreads();
    }
    if (tid == 0) sqPart[blockIdx.x] = red[0];
}

// ---------------------------------------------------------------------------
// Phase 2: single block, 16 waves. Each wave reduces 16-column tiles across
// the nblk partial rows using V_WMMA_F32_16X16X4_F32 with A = ones:
//   D[m][n] = sum_k B[k][n] + C[m][n]
// With A all-ones the K->lane/VGPR mapping of B is irrelevant (we sum every
// K slot), so each lane just loads two partial rows for its column.
// Then squares the resulting column sums, reduces, adds n*sqsum, finalizes.
// ---------------------------------------------------------------------------
__global__ void __launch_bounds__(TPB)
pel_finalize_kernel(const float* __restrict__ colPart,  // [nblk][2048]
                    const float* __restrict__ sqPart,   // [nblk]
                    float* __restrict__ out,
                    int n, int nblk) {
    const int tid   = threadIdx.x;       // 0..511  (16 waves)
    const int wave  = tid >> 5;
    const int lane  = tid & 31;
    const int laneN = lane & 15;         // column within 16-wide tile
    const int rsel  = lane >> 4;         // 0 for lanes 0-15, 1 for 16-31

    v2f ones;
    ones.x = 1.0f; ones.y = 1.0f;        // A-matrix (16x4) = all ones

    float acc = 0.f;                     // per-lane sum of colsum^2

    // 2048/16 = 128 tiles; 16 waves -> 8 tiles per wave (uniform, EXEC all-1s)
    for (int tile = wave; tile < D_COLS / 16; tile += 16) {
        const int cbase = tile * 16;
        v8f d = {0.f, 0.f, 0.f, 0.f, 0.f, 0.f, 0.f, 0.f};

        for (int b = 0; b < nblk; b += 4) {
            // B (4x16): lane supplies rows {b+rsel, b+2+rsel} for column cbase+laneN
            v2f bm;
            bm.x = colPart[(size_t)(b + rsel)     * D_COLS + cbase + laneN];
            bm.y = colPart[(size_t)(b + 2 + rsel) * D_COLS + cbase + laneN];
            d = __builtin_amdgcn_wmma_f32_16x16x4_f32(
                    /*neg_a=*/false, ones,
                    /*neg_b=*/false, bm,
                    /*c_mod=*/(short)0, d,
                    /*reuse_a=*/false, /*reuse_b=*/false);
        }
        // Every row of D equals the column sums; take M=0 row (lanes 0-15, d[0]).
        const float cs = d[0];
        if (lane < 16) acc += cs * cs;
    }

    __shared__ float red[TPB];
    red[tid] = acc;
    __syncthreads();
    for (int s = TPB / 2; s > 0; s >>= 1) {
        if (tid < s) red[tid] += red[tid + s];
        __syncthreads();
    }
    const float sumColSq = red[0];
    __syncthreads();

    // Reduce sqsum partials (nblk <= 256 values).
    float sqloc = 0.f;
    for (int i = tid; i < nblk; i += TPB) sqloc += sqPart[i];
    red[tid] = sqloc;
    __syncthreads();
    for (int s = TPB / 2; s > 0; s >>= 1) {
        if (tid < s) red[tid] += red[tid + s];
        __syncthreads();
    }

    if (tid == 0) {
        const double S      = (double)red[0];                  // sum ||x||^2
        const double total  = (double)n * S - (double)sumColSq;
        const double npairs = (double)n * (double)(n - 1) * 0.5;
        out[0] = (float)((total * 0.5) / npairs);
    }
}

// ---------------------------------------------------------------------------
extern "C" void kernel_launch(void* const* d_in, const int* in_sizes, int n_in,
                              void* d_out, int out_size, void* d_ws, size_t ws_size,
                              hipStream_t stream) {
    const float* X = (const float*)d_in[0];
    const int n = in_sizes[0] / D_COLS;   // 16384 rows

    // Pick block count that fits workspace; keep multiple of 4 for WMMA b-loop.
    int nblk = 256;
    while ((size_t)nblk * (D_COLS + 1) * sizeof(float) > ws_size && nblk > 4)
        nblk >>= 1;

    float* colPart = (float*)d_ws;                       // nblk * 2048 floats
    float* sqPart  = colPart + (size_t)nblk * D_COLS;    // nblk floats

    pel_colsum_sq_kernel<<<nblk, TPB, 0, stream>>>(X, colPart, sqPart, n, nblk);
    pel_finalize_kernel<<<1, TPB, 0, stream>>>(colPart, sqPart, (float*)d_out, n, nblk);
}